// VQWeightedAvgPool_17265768530685
// MI455X (gfx1250) — compile-verified
//
#include <hip/hip_runtime.h>

typedef __attribute__((ext_vector_type(2))) float v2f;
typedef __attribute__((ext_vector_type(8))) float v8f;

#define B_   32
#define NL_  13
#define L_   1024
#define D_   768
#define LCHUNK 256   // 4 chunks over L
#define DTILES (D_ / 16)   // 48

// ---------------------------------------------------------------------------
// Kernel 1: per-batch run-length weights + zero the output.
// One block of 1024 threads per batch (32 waves, wave32).
// ---------------------------------------------------------------------------
__global__ void vq_weights_kernel(const int* __restrict__ vq,       // (B, L, 2) int32
                                  const int* __restrict__ lengths,  // (B,)
                                  float* __restrict__ w,            // (B, L) workspace
                                  float* __restrict__ out)          // (B, D) -> zeroed
{
    const int b = blockIdx.x;
    const int l = threadIdx.x;  // 0..1023

    __shared__ int   scan[L_];
    __shared__ float runlen[L_];

    const int length = lengths[b];
    const bool valid = l < length;

    const int2* vqp = (const int2*)vq;  // (B, L) pairs
    int2 v = vqp[b * L_ + l];
    int2 vprev = (l > 0) ? vqp[b * L_ + l - 1] : make_int2(0, 0);
    const bool same = (l > 0) && (v.x == vprev.x) && (v.y == vprev.y);
    const bool boundary = valid && (l == 0 || !same);

    scan[l]   = boundary ? 1 : 0;
    runlen[l] = 0.0f;
    __syncthreads();

    // Inclusive Hillis-Steele scan over 1024 elements.
    for (int off = 1; off < L_; off <<= 1) {
        int add = (l >= off) ? scan[l - off] : 0;
        __syncthreads();
        scan[l] += add;
        __syncthreads();
    }

    int seg = scan[l] - 1;
    seg = (seg < 0) ? 0 : (seg > L_ - 1 ? L_ - 1 : seg);

    if (valid) atomicAdd(&runlen[seg], 1.0f);   // ds_add_f32
    __syncthreads();

    const float num_runs = (float)scan[L_ - 1];
    const float rl = fmaxf(runlen[seg], 1.0f);
    w[b * L_ + l] = valid ? (1.0f / (num_runs * rl)) : 0.0f;

    if (l < D_) out[b * D_ + l] = 0.0f;         // zero output (harness poisons it)
}

// ---------------------------------------------------------------------------
// Kernel 2: weighted reduction out[b,d] = sum_l w[b,l]*feat[b,l,d] via
// V_WMMA_F32_16X16X4_F32. Each wave: one 16-wide D tile x one 256-long L chunk.
//   A (16x4):  A[m,k] = w[l0+k] for every m  (lanes 0-15: K0/K1, 16-31: K2/K3)
//   B (4x16):  B[k,n] = feat[l0+k, d0+n]     (N striped across lanes)
// All 16 rows of D are identical; row 0 (c[0], N = lane&15) is the result.
// ---------------------------------------------------------------------------
__global__ void vq_wmma_reduce(const float* __restrict__ feat_base,  // (B, NL, L, D)
                               const float* __restrict__ w,          // (B, L)
                               float* __restrict__ out)              // (B, D)
{
    const int b        = blockIdx.x;
    const int lane     = threadIdx.x & 31;
    const int waveLoc  = threadIdx.x >> 5;            // 0..7
    const int waveGlob = blockIdx.y * 8 + waveLoc;    // 0..191
    const int dtile    = waveGlob % DTILES;           // 0..47
    const int chunk    = waveGlob / DTILES;           // 0..3
    const int d0       = dtile * 16;
    const int lbase    = chunk * LCHUNK;

    const float* __restrict__ feat =
        feat_base + (size_t)b * ((size_t)NL_ * L_ * D_) + (size_t)(NL_ - 1) * L_ * D_;
    const float* __restrict__ wb = w + b * L_;

    const int khi = (lane >> 4) << 1;   // 0 for lanes 0-15, 2 for lanes 16-31
    const int n   = lane & 15;

    v8f acc = {};
    #pragma unroll 4
    for (int l0 = lbase; l0 < lbase + LCHUNK; l0 += 4) {
        v2f a, bm;
        a.x  = wb[l0 + khi];
        a.y  = wb[l0 + khi + 1];
        bm.x = feat[(size_t)(l0 + khi)     * D_ + d0 + n];
        bm.y = feat[(size_t)(l0 + khi + 1) * D_ + d0 + n];
        acc = __builtin_amdgcn_wmma_f32_16x16x4_f32(
            /*neg_a=*/false, a, /*neg_b=*/false, bm,
            /*c_mod=*/(short)0, acc, /*reuse_a=*/false, /*reuse_b=*/false);
    }

    // c[0]: lanes 0-15 -> (M=0, N=lane); lanes 16-31 -> (M=8, N=lane-16).
    // Rows identical, so lanes 0-15 carry the full 16-wide result.
    if (lane < 16) atomicAdd(&out[b * D_ + d0 + n], acc[0]);
}

// ---------------------------------------------------------------------------
extern "C" void kernel_launch(void* const* d_in, const int* in_sizes, int n_in,
                              void* d_out, int out_size, void* d_ws, size_t ws_size,
                              hipStream_t stream) {
    const float* feat    = (const float*)d_in[0];  // (B, NL, L, D) f32
    const int*   lengths = (const int*)d_in[1];    // (B,)
    const int*   vq      = (const int*)d_in[2];    // (B, L, 2)
    float*       out     = (float*)d_out;          // (B, D) f32
    float*       w       = (float*)d_ws;           // (B, L) f32 = 128 KB scratch

    vq_weights_kernel<<<dim3(B_), dim3(L_), 0, stream>>>(vq, lengths, w, out);
    vq_wmma_reduce<<<dim3(B_, 24), dim3(256), 0, stream>>>(feat, w, out);
}